// InterpolatedLogPolarTransform_54485955117711
// MI455X (gfx1250) — compile-verified
//
#include <hip/hip_runtime.h>

// Problem constants (fixed by setup_inputs in the reference).
#define BATCH   8
#define FEAT    3
#define NPLANE  (BATCH * FEAT)   // 24
#define H       224
#define W       224
#define HS      112
#define WS      112
#define NA      12
#define NT      8
#define NK      (NA * NT)        // 96
#define RPB     8                // output rows per workgroup
#define WIN     136              // input-row window: 2*(RPB-1) + 122 rounded up
#define THREADS 224              // 7 wave32

// ---------------- compile-time offset / weight tables ----------------
static constexpr float cfloorf(float v) {
    int i = (int)v;                       // trunc toward zero
    return (float)((v < (float)i) ? i - 1 : i);
}

struct LPTab {
    float w00[NK], w01[NK], w10[NK], w11[NK];
    int   iy0[NK], ix0[NK];
    constexpr LPTab() : w00{}, w01{}, w10{}, w11{}, iy0{}, ix0{} {
        // q = 60^(1/7) via Newton on r^7 = 60
        double q = 2.0;
        for (int i = 0; i < 128; ++i) {
            double r2 = q * q, r3 = r2 * q, r6 = r3 * r3;
            q = q - (r6 * q - 60.0) / (7.0 * r6);
        }
        // sqrt(3)/2 via Newton
        double r = 1.7;
        for (int i = 0; i < 64; ++i) r = 0.5 * (r + 3.0 / r);
        const double s3h = 0.5 * r;
        // theta_a = a*pi/6 - pi  ->  sin = -sin(a*pi/6), cos = -cos(a*pi/6)
        const double sv[NA] = {0.0,-0.5,-s3h,-1.0,-s3h,-0.5, 0.0, 0.5, s3h, 1.0, s3h, 0.5};
        const double cv[NA] = {-1.0,-s3h,-0.5, 0.0, 0.5, s3h, 1.0, s3h, 0.5, 0.0,-0.5,-s3h};
        double taus[NT] = {};
        double tau = 1.0;
        for (int t = 0; t < NT; ++t) { taus[t] = tau; tau *= q; }
        for (int a = 0; a < NA; ++a)
            for (int t = 0; t < NT; ++t) {
                const int k = a * NT + t;
                const float yo = (float)(taus[t] * sv[a]);
                const float xo = (float)(taus[t] * cv[a]);
                const float fy = yo - cfloorf(yo);
                const float fx = xo - cfloorf(xo);
                iy0[k] = (int)cfloorf(yo);
                ix0[k] = (int)cfloorf(xo);
                w00[k] = (1.0f - fy) * (1.0f - fx);
                w01[k] = (1.0f - fy) * fx;
                w10[k] = fy * (1.0f - fx);
                w11[k] = fy * fx;
            }
    }
};
constexpr LPTab TAB{};

// ----------------------------- kernel --------------------------------
__global__ __launch_bounds__(THREADS)
void InterpolatedLogPolarTransform_kernel(const float* __restrict__ inp,
                                          float* __restrict__ out)
{
    extern __shared__ float smem[];            // WIN x W tile, rows zero-padded

    const int tid  = threadIdx.x;
    const int rb   = blockIdx.x;               // row block 0..13
    const int p    = blockIdx.y;               // plane 0..23  (b*FEAT + f)
    const int i0   = rb * RPB;
    const int ytop = 2 * i0 - 60;              // window row 0 == image row ytop

    // ---- zero the out-of-image rows (top/bottom halo only) ----
    for (int c = tid; c < WIN * (W / 4); c += THREADS) {
        const int rr = c / (W / 4);
        const int cc = (c - rr * (W / 4)) * 4;
        const int y  = ytop + rr;
        if ((unsigned)y >= (unsigned)H) {
            float4 z = {0.f, 0.f, 0.f, 0.f};
            *(float4*)&smem[rr * W + cc] = z;          // 16B aligned
        }
    }

    // ---- async-stage valid image rows into the window ----
    const int r0img = max(0, ytop);
    const int r1img = min(H, ytop + WIN);
    const int nvr   = r1img - r0img;
    const float* plane = inp + (size_t)p * (H * W);
    for (int c = tid; c < nvr * (W / 4); c += THREADS) {
        const int rr = c / (W / 4);
        const int cc = (c - rr * (W / 4)) * 4;
        const int y  = r0img + rr;
        const int lr = y - ytop;
        const float* g = plane + (size_t)y * W + cc;
        // Low 32 bits of a generic shared pointer == LDS byte offset.
        unsigned lofs = (unsigned)(size_t)(const void*)&smem[lr * W + cc];
        asm volatile("global_load_async_to_lds_b128 %0, %1, off"
                     :: "v"(lofs), "v"(g) : "memory");
    }
    asm volatile("s_wait_asynccnt 0" ::: "memory");
    __syncthreads();

    // ---- compute: thread owns column j, row parity rpar ----
    const int j    = tid % WS;
    const int rpar = tid / WS;                 // 0 or 1

    for (int k = 0; k < NK; ++k) {
        const int t  = k & 7, a = k >> 3;
        const int Y0 = TAB.iy0[k];
        const int X0 = TAB.ix0[k];

        // Column bounds folded into per-k weights (x is fixed per thread,k).
        const int  x   = 2 * j + X0;                   // in [-60, 283]
        const bool xv0 = (unsigned)x       < (unsigned)W;
        const bool xv1 = (unsigned)(x + 1) < (unsigned)W;
        const int  xc0 = xv0 ? x       : 0;
        const int  xc1 = xv1 ? (x + 1) : 0;
        const float mw00 = xv0 ? TAB.w00[k] : 0.f;
        const float mw01 = xv1 ? TAB.w01[k] : 0.f;
        const float mw10 = xv0 ? TAB.w10[k] : 0.f;
        const float mw11 = xv1 ? TAB.w11[k] : 0.f;

        const size_t kbase  = ((size_t)(p * NT + t) * NA + a) * (size_t)(HS * WS);
        const int    lybase = 2 * rpar + Y0 + 60;      // window row, always valid

        #pragma unroll
        for (int it = 0; it < RPB / 2; ++it) {
            const int ro = (lybase + 4 * it) * W;
            const float v00 = smem[ro + xc0];
            const float v01 = smem[ro + xc1];
            const float v10 = smem[ro + W + xc0];
            const float v11 = smem[ro + W + xc1];
            const float r = fmaf(mw00, v00,
                            fmaf(mw01, v01,
                            fmaf(mw10, v10, mw11 * v11)));
            const int i = i0 + 2 * it + rpar;
            __builtin_nontemporal_store(r, &out[kbase + (size_t)i * WS + j]);
        }
    }
}

extern "C" void kernel_launch(void* const* d_in, const int* in_sizes, int n_in,
                              void* d_out, int out_size, void* d_ws, size_t ws_size,
                              hipStream_t stream) {
    (void)in_sizes; (void)n_in; (void)d_ws; (void)ws_size; (void)out_size;
    const float* inp = (const float*)d_in[0];
    float* out = (float*)d_out;

    dim3 grid(HS / RPB, NPLANE);                 // 14 x 24 = 336 workgroups
    dim3 block(THREADS);                         // 224 threads = 7 waves
    size_t shmem = (size_t)(WIN * W) * sizeof(float);   // 121,856 B -> 2 WG/WGP
    InterpolatedLogPolarTransform_kernel<<<grid, block, shmem, stream>>>(inp, out);
}